// ImageLocalStdDev_19602230739813
// MI455X (gfx1250) — compile-verified
//
#include <hip/hip_runtime.h>
#include <math.h>

// ---------------------------------------------------------------------------
// 5x5 local std-dev, reflect padding, [16,3,1024,1024] fp32.
// Memory-bound (~402 MB traffic -> ~17us floor at 23.3 TB/s). Separable box
// filter through LDS. CDNA5 paths: async global->LDS fill (ASYNCcnt),
// b128 LDS/global vector ops, non-temporal output stores.
// ---------------------------------------------------------------------------

#define TX 64               // output tile width
#define TY 64               // output tile height
#define PAD 2               // 5x5 window -> halo of 2
#define PW (TX + 2 * PAD)   // 68 padded cols (row stride 272B = 17*16, aligned)
#define PH (TY + 2 * PAD)   // 68 padded rows
#define NTHREADS 256        // 8 waves (wave32)

typedef float f4 __attribute__((ext_vector_type(4)));

#if defined(__gfx1250__) && __has_builtin(__builtin_amdgcn_global_load_async_to_lds_b32)
#define USE_ASYNC_LDS 1
typedef __attribute__((address_space(1))) int* glob_i32_ptr;
typedef __attribute__((address_space(3))) int* lds_i32_ptr;
#else
#define USE_ASYNC_LDS 0
#endif

__device__ __forceinline__ int reflect_idx(int i, int n) {
    // JAX 'reflect' (mirror excluding edge). Only one bounce needed: pad << n.
    i = (i < 0) ? -i : i;
    i = (i >= n) ? (2 * n - 2 - i) : i;
    return i;
}

__global__ __launch_bounds__(NTHREADS)
void local_stddev5_kernel(const float* __restrict__ img,
                          float* __restrict__ out,
                          int H, int W) {
    __shared__ __align__(16) float sx [PH * PW];  // padded input tile
    __shared__ __align__(16) float rs [PH * TX];  // horizontal box-sum of x
    __shared__ __align__(16) float rs2[PH * TX];  // horizontal box-sum of x^2

    const int tid   = threadIdx.x;
    const int tileX = blockIdx.x * TX;
    const int tileY = blockIdx.y * TY;
    const long plane = (long)blockIdx.z * (long)H * (long)W;
    const float* __restrict__ src = img + plane;
    float* __restrict__ dst       = out + plane;

    // ---- Phase 1: fill padded tile (reflect at image borders) -------------
    for (int idx = tid; idx < PH * PW; idx += NTHREADS) {
        int r = idx / PW;
        int c = idx - r * PW;
        int gy = reflect_idx(tileY + r - PAD, H);
        int gx = reflect_idx(tileX + c - PAD, W);
        long goff = (long)gy * W + gx;
#if USE_ASYNC_LDS
        // each lane copies 4 bytes from its global address into its LDS slot, tracked by ASYNCcnt (no VGPR bounce).
        __builtin_amdgcn_global_load_async_to_lds_b32(
            (glob_i32_ptr)(src + goff),
            (lds_i32_ptr)(&sx[idx]),
            0, 0);
#else
        sx[idx] = src[goff];
#endif
    }
#if USE_ASYNC_LDS
#if __has_builtin(__builtin_amdgcn_s_wait_asynccnt)
    __builtin_amdgcn_s_wait_asynccnt(0);
#else
    asm volatile("s_wait_asynccnt 0x0" ::: "memory");
#endif
#endif
    __syncthreads();

    // ---- Phase 2: horizontal 5-wide sums of x and x^2, 4 cols/thread ------
    // Tasks: PH * (TX/4) = 68*16 = 1088.
    for (int t = tid; t < PH * (TX / 4); t += NTHREADS) {
        int r = t >> 4;                 // TX/4 == 16
        int c = (t & 15) << 2;          // output col, multiple of 4
        const float* row = &sx[r * PW + c];     // 16B aligned
        f4 a = *(const f4*)(row);               // padded cols c..c+3
        f4 b = *(const f4*)(row + 4);           // padded cols c+4..c+7
        float v[8] = {a.x, a.y, a.z, a.w, b.x, b.y, b.z, b.w};
        float q[8];
#pragma unroll
        for (int i = 0; i < 8; ++i) q[i] = v[i] * v[i];
        f4 S, S2;
#pragma unroll
        for (int j = 0; j < 4; ++j) {
            float s  = v[j] + v[j+1] + v[j+2] + v[j+3] + v[j+4];
            float s2 = q[j] + q[j+1] + q[j+2] + q[j+3] + q[j+4];
            S[j]  = s;
            S2[j] = s2;
        }
        *(f4*)&rs [r * TX + c] = S;     // 16B aligned (row stride 256B)
        *(f4*)&rs2[r * TX + c] = S2;
    }
    __syncthreads();

    // ---- Phase 3: vertical 5-wide sums + finalize, 4 cols/thread ----------
    // Tasks: TY * (TX/4) = 64*16 = 1024 -> exactly 4 iterations/thread.
    const float inv_n = 1.0f / 25.0f;
    for (int t = tid; t < TY * (TX / 4); t += NTHREADS) {
        int r = t >> 4;
        int c = (t & 15) << 2;
        f4 s  = {0.0f, 0.0f, 0.0f, 0.0f};
        f4 s2 = {0.0f, 0.0f, 0.0f, 0.0f};
#pragma unroll
        for (int k = 0; k < 5; ++k) {
            s  += *(const f4*)&rs [(r + k) * TX + c];
            s2 += *(const f4*)&rs2[(r + k) * TX + c];
        }
        f4 o;
#pragma unroll
        for (int j = 0; j < 4; ++j) {
            float mean = s[j] * inv_n;
            float var  = fmaf(-mean, mean, s2[j] * inv_n);  // E[x^2]-mean^2
            o[j] = sqrtf(var + 1e-9f);
        }
        // Output never re-read: non-temporal b128 store (16B aligned:
        // W=1024, tileX mult of 64, c mult of 4).
        __builtin_nontemporal_store(
            o, (f4*)&dst[(long)(tileY + r) * W + (tileX + c)]);
    }
}

extern "C" void kernel_launch(void* const* d_in, const int* in_sizes, int n_in,
                              void* d_out, int out_size, void* d_ws, size_t ws_size,
                              hipStream_t stream) {
    (void)in_sizes; (void)n_in; (void)d_ws; (void)ws_size; (void)out_size;
    const float* img = (const float*)d_in[0];
    float* out = (float*)d_out;

    const int B = 16, C = 3, H = 1024, W = 1024;
    dim3 grid(W / TX, H / TY, B * C);   // 16 x 16 x 48
    dim3 block(NTHREADS);
    hipLaunchKernelGGL(local_stddev5_kernel, grid, block, 0, stream,
                       img, out, H, W);
}